// ImprovedGAT_40999757807927
// MI455X (gfx1250) — compile-verified
//
#include <hip/hip_runtime.h>
#include <math.h>

typedef float v2f __attribute__((ext_vector_type(2)));
typedef float v8f __attribute__((ext_vector_type(8)));

#define NNODES 50000
#define NEDGES 800000
#define ETOT   (NEDGES + NNODES)
#define INC 128
#define HIDC 64
#define NH 4
#define OUTC 40
#define NLAYERS 4

// ---------- ordered-float encoding for atomicMax on u32 ----------
__device__ __forceinline__ unsigned ford(float f) {
  unsigned u = __float_as_uint(f);
  return (u & 0x80000000u) ? ~u : (u | 0x80000000u);
}
__device__ __forceinline__ float funord(unsigned u) {
  return (u & 0x80000000u) ? __uint_as_float(u & 0x7fffffffu) : __uint_as_float(~u);
}

// ---------- fp32 WMMA 16x16 tile: C = A[M x K] * B[K x N], row-major ----------
__device__ __forceinline__ v8f wmma_tile_f32(const float* __restrict__ A, int lda,
                                             const float* __restrict__ B, int ldb,
                                             int m0, int n0, int K) {
  const int lane = threadIdx.x & 31;
  const int r = lane & 15;
  const int kh = (lane >> 4) << 1;             // 0 for lanes 0-15, 2 for 16-31
  v8f c = {0.f, 0.f, 0.f, 0.f, 0.f, 0.f, 0.f, 0.f};
  for (int k = 0; k < K; k += 4) {
    v2f a, b;
    const float* ap = A + (size_t)(m0 + r) * lda + (k + kh);
    a.x = ap[0];
    a.y = ap[1];
    b.x = B[(size_t)(k + kh) * ldb + (n0 + r)];
    b.y = B[(size_t)(k + kh + 1) * ldb + (n0 + r)];
    c = __builtin_amdgcn_wmma_f32_16x16x4_f32(false, a, false, b, (short)0, c, false, false);
  }
  return c;
}

// ---------- h0 = elu(x @ Wi + bi) ----------
__global__ void gemm_input_elu(const float* __restrict__ x, const float* __restrict__ Wi,
                               const float* __restrict__ bi, float* __restrict__ h) {
  const int tiles_n = HIDC / 16;                       // 4
  const int tiles_m = NNODES / 16;                     // 3125
  int wave = blockIdx.x * (blockDim.x >> 5) + (threadIdx.x >> 5);
  if (wave >= tiles_m * tiles_n) return;
  int tm = wave / tiles_n, tn = wave % tiles_n;
  v8f c = wmma_tile_f32(x, INC, Wi, HIDC, tm * 16, tn * 16, INC);
  int lane = threadIdx.x & 31;
  int col = tn * 16 + (lane & 15);
  int row0 = tm * 16 + ((lane >> 4) << 3);
  float bias = bi[col];
  for (int r = 0; r < 8; ++r) {
    float v = c[r] + bias;
    v = v > 0.f ? v : expm1f(v);
    h[(size_t)(row0 + r) * HIDC + col] = v;
  }
}

// ---------- xl = h @ W  (N x 64) @ (64 x 256) ----------
__global__ void gemm_layer(const float* __restrict__ h, const float* __restrict__ W,
                           float* __restrict__ xl) {
  const int tiles_n = (NH * HIDC) / 16;                // 16
  const int tiles_m = NNODES / 16;
  int wave = blockIdx.x * (blockDim.x >> 5) + (threadIdx.x >> 5);
  if (wave >= tiles_m * tiles_n) return;
  int tm = wave / tiles_n, tn = wave % tiles_n;
  v8f c = wmma_tile_f32(h, HIDC, W, NH * HIDC, tm * 16, tn * 16, HIDC);
  int lane = threadIdx.x & 31;
  int col = tn * 16 + (lane & 15);
  int row0 = tm * 16 + ((lane >> 4) << 3);
  for (int r = 0; r < 8; ++r)
    xl[(size_t)(row0 + r) * (NH * HIDC) + col] = c[r];
}

// ---------- out = h @ Wo + bo  (OUT=40, guarded) ----------
__global__ void gemm_out_kernel(const float* __restrict__ h, const float* __restrict__ Wo,
                                const float* __restrict__ bo, float* __restrict__ out) {
  const int tiles_n = 3;                               // ceil(40/16)
  const int tiles_m = NNODES / 16;
  int wave = blockIdx.x * (blockDim.x >> 5) + (threadIdx.x >> 5);
  if (wave >= tiles_m * tiles_n) return;
  int tm = wave / tiles_n, tn = wave % tiles_n;
  const int lane = threadIdx.x & 31;
  const int r16 = lane & 15;
  const int kh = (lane >> 4) << 1;
  int col = tn * 16 + r16;
  bool ok = col < OUTC;
  v8f c = {0.f, 0.f, 0.f, 0.f, 0.f, 0.f, 0.f, 0.f};
  for (int k = 0; k < HIDC; k += 4) {
    v2f a, b;
    const float* ap = h + (size_t)(tm * 16 + r16) * HIDC + (k + kh);
    a.x = ap[0];
    a.y = ap[1];
    b.x = ok ? Wo[(size_t)(k + kh) * OUTC + col] : 0.f;
    b.y = ok ? Wo[(size_t)(k + kh + 1) * OUTC + col] : 0.f;
    c = __builtin_amdgcn_wmma_f32_16x16x4_f32(false, a, false, b, (short)0, c, false, false);
  }
  if (ok) {
    float bias = bo[col];
    int row0 = tm * 16 + ((lane >> 4) << 3);
    for (int r = 0; r < 8; ++r)
      out[(size_t)(row0 + r) * OUTC + col] = c[r] + bias;
  }
}

// ---------- per-node attention logits ----------
__global__ void node_alpha_kernel(const float* __restrict__ xl, const float* __restrict__ aS,
                                  const float* __restrict__ aD, float* __restrict__ alphaS,
                                  float* __restrict__ alphaD) {
  int t = blockIdx.x * blockDim.x + threadIdx.x;
  if (t >= NNODES * NH) return;
  int n = t >> 2, hh = t & 3;
  const float* v = xl + (size_t)n * (NH * HIDC) + hh * HIDC;
  const float* s = aS + hh * HIDC;
  const float* d = aD + hh * HIDC;
  float ss = 0.f, sd = 0.f;
  for (int c = 0; c < HIDC; ++c) { float x = v[c]; ss += x * s[c]; sd += x * d[c]; }
  alphaS[t] = ss;
  alphaD[t] = sd;
}

// ---------- per-layer init ----------
__global__ void init_layer_kernel(unsigned* __restrict__ mkey, float* __restrict__ denom,
                                  float* __restrict__ acc) {
  int t = blockIdx.x * blockDim.x + threadIdx.x;
  if (t < NNODES * NH) { mkey[t] = ford(-3.0e38f); denom[t] = 0.f; }
  if (t < NNODES * HIDC) acc[t] = 0.f;
}

// ---------- edge pass 1: leaky-relu logits + segment max ----------
__global__ void edge_max_kernel(const int* __restrict__ ei, const float* __restrict__ aS,
                                const float* __restrict__ aD, float* __restrict__ ebuf,
                                unsigned* __restrict__ mkey) {
  int e = blockIdx.x * blockDim.x + threadIdx.x;
  if (e >= ETOT) return;
  int s, d;
  if (e < NEDGES) { s = ei[e]; d = ei[NEDGES + e]; } else { s = d = e - NEDGES; }
  float4 vs = *reinterpret_cast<const float4*>(aS + (size_t)s * NH);
  float4 vd = *reinterpret_cast<const float4*>(aD + (size_t)d * NH);
  float eh[NH] = {vs.x + vd.x, vs.y + vd.y, vs.z + vd.z, vs.w + vd.w};
  float4 o;
  float* op = &o.x;
  for (int h = 0; h < NH; ++h) {
    float v = eh[h];
    v = v > 0.f ? v : 0.2f * v;
    op[h] = v;
    atomicMax(&mkey[(size_t)d * NH + h], ford(v));
  }
  *reinterpret_cast<float4*>(ebuf + (size_t)e * NH) = o;
}

// ---------- edge pass 2: p = exp(e - m[dst]); denom += p ----------
__global__ void edge_exp_kernel(const int* __restrict__ ei, float* __restrict__ ebuf,
                                const unsigned* __restrict__ mkey, float* __restrict__ denom) {
  int e = blockIdx.x * blockDim.x + threadIdx.x;
  if (e >= ETOT) return;
  int d = (e < NEDGES) ? ei[NEDGES + e] : (e - NEDGES);
  float4 ev = *reinterpret_cast<const float4*>(ebuf + (size_t)e * NH);
  uint4 mk = *reinterpret_cast<const uint4*>(mkey + (size_t)d * NH);
  float4 p;
  p.x = expf(ev.x - funord(mk.x));
  p.y = expf(ev.y - funord(mk.y));
  p.z = expf(ev.z - funord(mk.z));
  p.w = expf(ev.w - funord(mk.w));
  atomicAdd(&denom[(size_t)d * NH + 0], p.x);
  atomicAdd(&denom[(size_t)d * NH + 1], p.y);
  atomicAdd(&denom[(size_t)d * NH + 2], p.z);
  atomicAdd(&denom[(size_t)d * NH + 3], p.w);
  *reinterpret_cast<float4*>(ebuf + (size_t)e * NH) = p;
}

// ---------- edge pass 3: wave per edge, head-mean folded into accumulate ----------
__global__ void edge_agg_kernel(const int* __restrict__ ei, const float* __restrict__ xl,
                                const float* __restrict__ ebuf, const float* __restrict__ denom,
                                float* __restrict__ acc) {
  int t = blockIdx.x * blockDim.x + threadIdx.x;
  int e = t >> 5;
  int lane = t & 31;
  if (e >= ETOT) return;
  int s, d;
  if (e < NEDGES) { s = ei[e]; d = ei[NEDGES + e]; } else { s = d = e - NEDGES; }
  float4 p = *reinterpret_cast<const float4*>(ebuf + (size_t)e * NH);
  float4 dn = *reinterpret_cast<const float4*>(denom + (size_t)d * NH);
  float w0 = p.x / (dn.x + 1e-16f) * 0.25f;
  float w1 = p.y / (dn.y + 1e-16f) * 0.25f;
  float w2 = p.z / (dn.z + 1e-16f) * 0.25f;
  float w3 = p.w / (dn.w + 1e-16f) * 0.25f;
  const float* xs = xl + (size_t)s * (NH * HIDC);
  float* ap = acc + (size_t)d * HIDC;
  for (int half = 0; half < 2; ++half) {
    int c = lane + half * 32;
    float v = w0 * xs[c] + w1 * xs[HIDC + c] + w2 * xs[2 * HIDC + c] + w3 * xs[3 * HIDC + c];
    atomicAdd(&ap[c], v);
  }
}

// ---------- bias + layernorm + elu + residual, one wave32 per node ----------
__global__ void node_post_kernel(const float* __restrict__ acc, const float* __restrict__ hin,
                                 const float* __restrict__ gb, const float* __restrict__ lg,
                                 const float* __restrict__ lb, float* __restrict__ hout,
                                 int add_res) {
  int t = blockIdx.x * blockDim.x + threadIdx.x;
  int n = t >> 5;
  int lane = t & 31;
  if (n >= NNODES) return;
  size_t base = (size_t)n * HIDC;
  float v0 = acc[base + lane] + gb[lane];
  float v1 = acc[base + 32 + lane] + gb[32 + lane];
  float s = v0 + v1;
  for (int m = 16; m > 0; m >>= 1) s += __shfl_xor(s, m, 32);
  float mu = s * (1.0f / HIDC);
  float d0 = v0 - mu, d1 = v1 - mu;
  float q = d0 * d0 + d1 * d1;
  for (int m = 16; m > 0; m >>= 1) q += __shfl_xor(q, m, 32);
  float inv = rsqrtf(q * (1.0f / HIDC) + 1e-5f);
  float y0 = d0 * inv * lg[lane] + lb[lane];
  float y1 = d1 * inv * lg[32 + lane] + lb[32 + lane];
  y0 = y0 > 0.f ? y0 : expm1f(y0);
  y1 = y1 > 0.f ? y1 : expm1f(y1);
  if (add_res) { y0 += hin[base + lane]; y1 += hin[base + 32 + lane]; }
  hout[base + lane] = y0;
  hout[base + 32 + lane] = y1;
}

extern "C" void kernel_launch(void* const* d_in, const int* in_sizes, int n_in,
                              void* d_out, int out_size, void* d_ws, size_t ws_size,
                              hipStream_t stream) {
  (void)in_sizes; (void)n_in; (void)out_size; (void)ws_size;
  const float* x    = (const float*)d_in[0];
  const int*   ei   = (const int*)d_in[1];
  const float* Wi   = (const float*)d_in[2];
  const float* bi   = (const float*)d_in[3];
  const float* linW = (const float*)d_in[4];
  const float* attS = (const float*)d_in[5];
  const float* attD = (const float*)d_in[6];
  const float* gatB = (const float*)d_in[7];
  const float* lnG  = (const float*)d_in[8];
  const float* lnB  = (const float*)d_in[9];
  const float* Wo   = (const float*)d_in[10];
  const float* bo   = (const float*)d_in[11];

  float* hA   = (float*)d_ws;
  float* hB   = hA + (size_t)NNODES * HIDC;
  float* xl   = hB + (size_t)NNODES * HIDC;
  float* aS   = xl + (size_t)NNODES * NH * HIDC;
  float* aD   = aS + (size_t)NNODES * NH;
  unsigned* mkey = (unsigned*)(aD + (size_t)NNODES * NH);
  float* denom = (float*)(mkey + (size_t)NNODES * NH);
  float* acc  = denom + (size_t)NNODES * NH;
  float* ebuf = acc + (size_t)NNODES * HIDC;

  dim3 blk(256);

  {
    int waves = (NNODES / 16) * (HIDC / 16);
    gemm_input_elu<<<(waves + 7) / 8, blk, 0, stream>>>(x, Wi, bi, hA);
  }

  float* hin = hA;
  float* hout = hB;
  for (int l = 0; l < NLAYERS; ++l) {
    {
      int waves = (NNODES / 16) * ((NH * HIDC) / 16);
      gemm_layer<<<(waves + 7) / 8, blk, 0, stream>>>(
          hin, linW + (size_t)l * HIDC * NH * HIDC, xl);
    }
    {
      int th = NNODES * NH;
      node_alpha_kernel<<<(th + 255) / 256, blk, 0, stream>>>(
          xl, attS + (size_t)l * NH * HIDC, attD + (size_t)l * NH * HIDC, aS, aD);
    }
    {
      int th = NNODES * HIDC;
      init_layer_kernel<<<(th + 255) / 256, blk, 0, stream>>>(mkey, denom, acc);
    }
    edge_max_kernel<<<(ETOT + 255) / 256, blk, 0, stream>>>(ei, aS, aD, ebuf, mkey);
    edge_exp_kernel<<<(ETOT + 255) / 256, blk, 0, stream>>>(ei, ebuf, mkey, denom);
    {
      long long th = (long long)ETOT * 32;
      edge_agg_kernel<<<(int)((th + 255) / 256), blk, 0, stream>>>(ei, xl, ebuf, denom, acc);
    }
    {
      int th = NNODES * 32;
      node_post_kernel<<<(th + 255) / 256, blk, 0, stream>>>(
          acc, hin, gatB + (size_t)l * HIDC, lnG + (size_t)l * HIDC,
          lnB + (size_t)l * HIDC, hout, l > 0 ? 1 : 0);
    }
    float* tmp = hin; hin = hout; hout = tmp;
  }

  {
    int waves = (NNODES / 16) * 3;
    gemm_out_kernel<<<(waves + 7) / 8, blk, 0, stream>>>(hin, Wo, bo, (float*)d_out);
  }
}